// MDTA_Attention_6511170421393
// MI455X (gfx1250) — compile-verified
//
#include <hip/hip_runtime.h>
#include <hip/hip_bf16.h>
#include <math.h>

// MDTA attention for MI455X (gfx1250, wave32, WMMA).
// All GEMM-like stages use V_WMMA_F32_16X16X4_F32 (fp32-exact; workload is
// HBM-bound at 23.3 TB/s so the fp32 matrix path loses nothing vs bf16).
// av_gemm additionally stages the attention matrix in LDS via the CDNA5
// async global->LDS path (ASYNCcnt).

typedef __attribute__((ext_vector_type(2))) float v2f;
typedef __attribute__((ext_vector_type(8))) float v8f;

#define BATCH 8
#define CDIM  192
#define C2DIM 384
#define HEADS 4
#define CH    48
#define HDIM  128
#define HW    16384

// ---------------------------------------------------------------------------
// Generic 1x1-conv GEMM:  Y[b][M][HW] = W[M][K] * X[b][K][HW]
// One wave computes a 16x64 strip (4 accumulators) so the A (weight) pair is
// loaded once per K-step and reused by 4 WMMAs. grid = (HW/64, M/16, BATCH).
// ---------------------------------------------------------------------------
__global__ __launch_bounds__(32) void gemm1x1_kernel(
    const float* __restrict__ W, const float* __restrict__ X,
    float* __restrict__ Y, int M, int K)
{
    const int lane   = threadIdx.x & 31;
    const int n_base = blockIdx.x * 64;
    const int m_base = blockIdx.y * 16;
    const int b      = blockIdx.z;

    const float* Xb = X + (size_t)b * K * HW;
    float*       Yb = Y + (size_t)b * M * HW;

    const int col   = lane & 15;        // A row index / B,C,D column index
    const int khalf = (lane >> 4) * 2;  // K sub-offset within the 16x16x4 op

    v8f acc0 = {}, acc1 = {}, acc2 = {}, acc3 = {};
    for (int k0 = 0; k0 < K; k0 += 4) {
        // A 16x4: lane<16 -> K={k0,k0+1}, lane>=16 -> K={k0+2,k0+3}.
        v2f a = *(const v2f*)(W + (size_t)(m_base + col) * K + k0 + khalf);
        const float* r0 = Xb + (size_t)(k0 + khalf)     * HW + n_base + col;
        const float* r1 = Xb + (size_t)(k0 + khalf + 1) * HW + n_base + col;
        v2f b0; b0.x = r0[0];  b0.y = r1[0];
        v2f b1; b1.x = r0[16]; b1.y = r1[16];
        v2f b2; b2.x = r0[32]; b2.y = r1[32];
        v2f b3; b3.x = r0[48]; b3.y = r1[48];
        acc0 = __builtin_amdgcn_wmma_f32_16x16x4_f32(false, a, false, b0, (short)0, acc0, false, false);
        acc1 = __builtin_amdgcn_wmma_f32_16x16x4_f32(false, a, false, b1, (short)0, acc1, false, false);
        acc2 = __builtin_amdgcn_wmma_f32_16x16x4_f32(false, a, false, b2, (short)0, acc2, false, false);
        acc3 = __builtin_amdgcn_wmma_f32_16x16x4_f32(false, a, false, b3, (short)0, acc3, false, false);
    }
    const int mo = (lane >> 4) * 8;   // C/D layout: VGPR r -> rows r and r+8
#pragma unroll
    for (int r = 0; r < 8; ++r) {
        float* yr = Yb + (size_t)(m_base + r + mo) * HW + n_base + col;
        yr[0]  = acc0[r];
        yr[16] = acc1[r];
        yr[32] = acc2[r];
        yr[48] = acc3[r];
    }
}

// ---------------------------------------------------------------------------
// Dense 3x3 conv as a 9-tap GEMM (q branch: 192->192, pad 1), W [M][K][3][3].
// Same 16x64-per-wave blocking: the tap weight pair feeds 4 WMMAs.
// grid = (HW/64, M/16, BATCH).
// ---------------------------------------------------------------------------
__global__ __launch_bounds__(32) void conv3x3_kernel(
    const float* __restrict__ W, const float* __restrict__ X,
    float* __restrict__ Y, int M, int K)
{
    const int lane   = threadIdx.x & 31;
    const int n_base = blockIdx.x * 64;
    const int m_base = blockIdx.y * 16;
    const int b      = blockIdx.z;

    const float* Xb = X + (size_t)b * K * HW;
    float*       Yb = Y + (size_t)b * M * HW;

    const int col   = lane & 15;
    const int khalf = (lane >> 4) * 2;

    int py[4], px[4];
#pragma unroll
    for (int t = 0; t < 4; ++t) {
        const int p = n_base + t * 16 + col;
        py[t] = p >> 7;
        px[t] = p & 127;
    }

    v8f acc0 = {}, acc1 = {}, acc2 = {}, acc3 = {};
    for (int tap = 0; tap < 9; ++tap) {
        const int dy = tap / 3 - 1, dx = tap % 3 - 1;
        int  qp[4];
        bool ok[4];
#pragma unroll
        for (int t = 0; t < 4; ++t) {
            const int qy = py[t] + dy, qx = px[t] + dx;
            ok[t] = (qy >= 0) && (qy < HDIM) && (qx >= 0) && (qx < HDIM);
            qp[t] = qy * HDIM + qx;
        }
        for (int k0 = 0; k0 < K; k0 += 4) {
            v2f a;
            a.x = W[((size_t)(m_base + col) * K + k0 + khalf)     * 9 + tap];
            a.y = W[((size_t)(m_base + col) * K + k0 + khalf + 1) * 9 + tap];
            const float* r0 = Xb + (size_t)(k0 + khalf)     * HW;
            const float* r1 = Xb + (size_t)(k0 + khalf + 1) * HW;
            v2f b0; b0.x = ok[0] ? r0[qp[0]] : 0.0f; b0.y = ok[0] ? r1[qp[0]] : 0.0f;
            v2f b1; b1.x = ok[1] ? r0[qp[1]] : 0.0f; b1.y = ok[1] ? r1[qp[1]] : 0.0f;
            v2f b2; b2.x = ok[2] ? r0[qp[2]] : 0.0f; b2.y = ok[2] ? r1[qp[2]] : 0.0f;
            v2f b3; b3.x = ok[3] ? r0[qp[3]] : 0.0f; b3.y = ok[3] ? r1[qp[3]] : 0.0f;
            acc0 = __builtin_amdgcn_wmma_f32_16x16x4_f32(false, a, false, b0, (short)0, acc0, false, false);
            acc1 = __builtin_amdgcn_wmma_f32_16x16x4_f32(false, a, false, b1, (short)0, acc1, false, false);
            acc2 = __builtin_amdgcn_wmma_f32_16x16x4_f32(false, a, false, b2, (short)0, acc2, false, false);
            acc3 = __builtin_amdgcn_wmma_f32_16x16x4_f32(false, a, false, b3, (short)0, acc3, false, false);
        }
    }
    const int mo = (lane >> 4) * 8;
#pragma unroll
    for (int r = 0; r < 8; ++r) {
        float* yr = Yb + (size_t)(m_base + r + mo) * HW + n_base + col;
        yr[0]  = acc0[r];
        yr[16] = acc1[r];
        yr[32] = acc2[r];
        yr[48] = acc3[r];
    }
}

// ---------------------------------------------------------------------------
// Depthwise 3x3 conv (kv branch, groups = 384). One thread per pixel.
// ---------------------------------------------------------------------------
__global__ __launch_bounds__(256) void dwconv3_kernel(
    const float* __restrict__ X, const float* __restrict__ Wd,
    float* __restrict__ Y, int Ctot)
{
    const int p = blockIdx.x * 256 + threadIdx.x;
    const int c = blockIdx.y;
    const int b = blockIdx.z;
    const float* Xc = X + ((size_t)b * Ctot + c) * HW;
    float*       Yc = Y + ((size_t)b * Ctot + c) * HW;
    const int py = p >> 7, px = p & 127;
    float s = 0.0f;
#pragma unroll
    for (int tap = 0; tap < 9; ++tap) {
        const int qy = py + tap / 3 - 1;
        const int qx = px + tap % 3 - 1;
        const bool valid = (qy >= 0) && (qy < HDIM) && (qx >= 0) && (qx < HDIM);
        const float v = valid ? Xc[qy * HDIM + qx] : 0.0f;
        s = fmaf(Wd[c * 9 + tap], v, s);
    }
    Yc[p] = s;
}

// ---------------------------------------------------------------------------
// L2-norm reciprocals per (b, channel) row: R[b*192+c] = 1/max(||row||, 1e-12)
// ---------------------------------------------------------------------------
__global__ __launch_bounds__(256) void rownorm_kernel(
    const float* __restrict__ X, float* __restrict__ R, int Ctot)
{
    const int c = blockIdx.x;     // 0..191 (k uses first 192 of kv's 384)
    const int b = blockIdx.y;
    const float* row = X + ((size_t)b * Ctot + c) * HW;
    float s = 0.0f;
    for (int i = threadIdx.x; i < HW; i += 256) {
        const float v = row[i];
        s = fmaf(v, v, s);
    }
    __shared__ float sm[256];
    sm[threadIdx.x] = s;
    __syncthreads();
    for (int o = 128; o > 0; o >>= 1) {
        if (threadIdx.x < o) sm[threadIdx.x] += sm[threadIdx.x + o];
        __syncthreads();
    }
    if (threadIdx.x == 0) {
        const float n = sqrtf(sm[0]);
        R[b * CDIM + c] = 1.0f / fmaxf(n, 1e-12f);
    }
}

// ---------------------------------------------------------------------------
// attn[c][d] = sum_n q[c][n]*k[d][n] * rq[c]*rk[d]   (per b,head; 48x48, K=16384)
// Both operands contiguous along K -> float2 loads + prefetch down the stream.
// grid = (3, 3, BATCH*HEADS), one wave per 16x16 tile.
// ---------------------------------------------------------------------------
__global__ __launch_bounds__(32) void attn_gemm_kernel(
    const float* __restrict__ Q, const float* __restrict__ KV,
    const float* __restrict__ RQ, const float* __restrict__ RK,
    float* __restrict__ ATT)
{
    const int lane   = threadIdx.x & 31;
    const int d_base = blockIdx.x * 16;   // N dim: k-channel d
    const int c_base = blockIdx.y * 16;   // M dim: q-channel c
    const int bh     = blockIdx.z;
    const int b = bh >> 2, h = bh & 3;

    const float* Qb = Q  + ((size_t)b * CDIM  + h * CH) * HW;
    const float* Kb = KV + ((size_t)b * C2DIM + h * CH) * HW;

    const int col   = lane & 15;
    const int khalf = (lane >> 4) * 2;

    const float* qrow = Qb + (size_t)(c_base + col) * HW;
    const float* krow = Kb + (size_t)(d_base + col) * HW;

    v8f acc = {};
    for (int k0 = 0; k0 < HW; k0 += 4) {
        if ((k0 & 255) == 0) {   // prefetch next L2 chunk of both streams
            __builtin_prefetch(qrow + k0 + 256, 0, 1);
            __builtin_prefetch(krow + k0 + 256, 0, 1);
        }
        v2f a  = *(const v2f*)(qrow + k0 + khalf);
        v2f bb = *(const v2f*)(krow + k0 + khalf);
        acc = __builtin_amdgcn_wmma_f32_16x16x4_f32(false, a, false, bb,
                                                    (short)0, acc, false, false);
    }
    const float* rq = RQ + b * CDIM + h * CH;
    const float* rk = RK + b * CDIM + h * CH;
    float* att = ATT + (size_t)bh * CH * CH;
    const int mo = (lane >> 4) * 8;
#pragma unroll
    for (int r = 0; r < 8; ++r) {
        const int c = c_base + r + mo;
        const int d = d_base + col;
        att[c * CH + d] = acc[r] * rq[c] * rk[d];
    }
}

// ---------------------------------------------------------------------------
// Row softmax over 48 entries with temperature*log(48). One thread per row.
// ---------------------------------------------------------------------------
__global__ __launch_bounds__(256) void softmax_kernel(
    float* __restrict__ ATT, const float* __restrict__ temperature)
{
    const int row = blockIdx.x * 256 + threadIdx.x;   // 0 .. B*HEADS*CH-1
    if (row >= BATCH * HEADS * CH) return;
    const int h = (row / CH) & 3;
    const float t = temperature[h] * logf(48.0f);
    float* a = ATT + (size_t)row * CH;
    float mx = -INFINITY;
    for (int i = 0; i < CH; ++i) mx = fmaxf(mx, a[i] * t);
    float s = 0.0f;
    for (int i = 0; i < CH; ++i) {
        const float e = expf(a[i] * t - mx);
        a[i] = e;
        s += e;
    }
    const float inv = 1.0f / s;
    for (int i = 0; i < CH; ++i) a[i] *= inv;
}

// ---------------------------------------------------------------------------
// out[c][n] = sum_d attn[c][d] * v[d][n]    (per b,head; M=48, K=48, N=16384)
// The 48x48 attn matrix is staged into LDS ONCE per workgroup with the CDNA5
// async global->LDS path (ASYNCcnt), then 6 waves (3 c-tiles x 2 n-tiles)
// read A operands from LDS. grid = (HW/32, 1, BATCH*HEADS), block = 192.
// ---------------------------------------------------------------------------
__global__ __launch_bounds__(192) void av_gemm_kernel(
    const float* __restrict__ ATT, const float* __restrict__ KV,
    float* __restrict__ O)
{
    __shared__ float satt[CH * CH];   // 48*48 floats = 9216 B

    const int tid  = threadIdx.x;
    const int lane = tid & 31;
    const int wv   = tid >> 5;            // 0..5
    const int bh   = blockIdx.z;
    const int b = bh >> 2, h = bh & 3;

    const float* att = ATT + (size_t)bh * CH * CH;   // uniform per block

    // Async-stage attn into LDS: each lane copies strided elements.
    // the LDS address comes from the VDST VGPR; generic->32-bit truncation of a shared
    // pointer yields the workgroup-relative LDS byte offset.
    for (int i = tid; i < CH * CH; i += 192) {
        unsigned ldsoff = (unsigned)(uintptr_t)(&satt[i]);
        int      voff   = i * 4;
        asm volatile("global_load_async_to_lds_b32 %0, %1, %2"
                     :: "v"(ldsoff), "v"(voff), "s"(att)
                     : "memory");
    }
    asm volatile("s_wait_asynccnt 0" ::: "memory");
    __syncthreads();

    const int c_base = (wv % 3) * 16;                    // 0,16,32
    const int n_base = blockIdx.x * 32 + (wv / 3) * 16;  // 2 n-tiles per block

    const float* Vb = KV + ((size_t)b * C2DIM + CDIM + h * CH) * HW;
    float*       Ob = O  + ((size_t)b * CDIM + h * CH) * HW;

    const int col   = lane & 15;
    const int khalf = (lane >> 4) * 2;

    v8f acc = {};
#pragma unroll
    for (int k0 = 0; k0 < CH; k0 += 4) {
        v2f a = *(const v2f*)(&satt[(c_base + col) * CH + k0 + khalf]);  // ds_load_b64
        v2f bb;
        bb.x = Vb[(size_t)(k0 + khalf)     * HW + n_base + col];
        bb.y = Vb[(size_t)(k0 + khalf + 1) * HW + n_base + col];
        acc = __builtin_amdgcn_wmma_f32_16x16x4_f32(false, a, false, bb,
                                                    (short)0, acc, false, false);
    }
    const int mo = (lane >> 4) * 8;
#pragma unroll
    for (int r = 0; r < 8; ++r)
        Ob[(size_t)(c_base + r + mo) * HW + n_base + col] = acc[r];
}

// ---------------------------------------------------------------------------
extern "C" void kernel_launch(void* const* d_in, const int* in_sizes, int n_in,
                              void* d_out, int out_size, void* d_ws, size_t ws_size,
                              hipStream_t stream)
{
    (void)in_sizes; (void)n_in; (void)out_size; (void)ws_size;

    const float* x       = (const float*)d_in[0];
    const float* y       = (const float*)d_in[1];
    const float* kv_w    = (const float*)d_in[2];   // [384,192]
    const float* kv_dw_w = (const float*)d_in[3];   // [384,1,3,3]
    const float* q_w     = (const float*)d_in[4];   // [192,192]
    const float* q_dw_w  = (const float*)d_in[5];   // [192,192,3,3]
    const float* proj_w  = (const float*)d_in[6];   // [192,192]
    const float* temper  = (const float*)d_in[7];   // [4]
    float* out = (float*)d_out;

    // Workspace layout (floats). buf1 is reused 3x (kv1 -> q1 -> attn_out).
    const size_t SZ_KV = (size_t)BATCH * C2DIM * HW;   // 50,331,648
    const size_t SZ_Q  = (size_t)BATCH * CDIM  * HW;   // 25,165,824
    float* buf1 = (float*)d_ws;            // kv1 / q1 / attn_out
    float* buf2 = buf1 + SZ_KV;            // kv (post depthwise)
    float* buf3 = buf2 + SZ_KV;            // q (post dense 3x3)
    float* rq   = buf3 + SZ_Q;             // [8*192]
    float* rk   = rq + BATCH * CDIM;       // [8*192]
    float* attn = rk + BATCH * CDIM;       // [8*4*48*48]

    // 1) kv 1x1:  buf1 = kv_w @ x          (M=384, K=192)
    gemm1x1_kernel<<<dim3(HW / 64, C2DIM / 16, BATCH), 32, 0, stream>>>(
        kv_w, x, buf1, C2DIM, CDIM);
    // 2) kv depthwise 3x3:  buf2 = dw(buf1)
    dwconv3_kernel<<<dim3(HW / 256, C2DIM, BATCH), 256, 0, stream>>>(
        buf1, kv_dw_w, buf2, C2DIM);
    // 3) q 1x1:  buf1 = q_w @ y            (M=192, K=192) — reuses buf1
    gemm1x1_kernel<<<dim3(HW / 64, CDIM / 16, BATCH), 32, 0, stream>>>(
        q_w, y, buf1, CDIM, CDIM);
    // 4) q dense 3x3:  buf3 = q_dw_w *conv* buf1   (the 87-GFLOP stage)
    conv3x3_kernel<<<dim3(HW / 64, CDIM / 16, BATCH), 32, 0, stream>>>(
        q_dw_w, buf1, buf3, CDIM, CDIM);
    // 5) l2-norm reciprocals for q (buf3) and k (= first 192 chans of buf2)
    rownorm_kernel<<<dim3(CDIM, BATCH), 256, 0, stream>>>(buf3, rq, CDIM);
    rownorm_kernel<<<dim3(CDIM, BATCH), 256, 0, stream>>>(buf2, rk, C2DIM);
    // 6) attn = (q k^T) * rq rk ; softmax with temperature*log(48)
    attn_gemm_kernel<<<dim3(3, 3, BATCH * HEADS), 32, 0, stream>>>(
        buf3, buf2, rq, rk, attn);
    softmax_kernel<<<dim3(6), 256, 0, stream>>>(attn, temper);
    // 7) out = attn @ v   -> buf1 (reused again); attn staged via async-LDS
    av_gemm_kernel<<<dim3(HW / 32, 1, BATCH * HEADS), 192, 0, stream>>>(
        attn, buf2, buf1);
    // 8) proj 1x1: d_out = proj_w @ buf1
    gemm1x1_kernel<<<dim3(HW / 64, CDIM / 16, BATCH), 32, 0, stream>>>(
        proj_w, buf1, out, CDIM, CDIM);
}